// GATNet_81415400063709
// MI455X (gfx1250) — compile-verified
//
#include <hip/hip_runtime.h>

#define NEG_SLOPE 0.2f
#define EPS_F 1e-16f

typedef __attribute__((ext_vector_type(2))) float v2f;
typedef __attribute__((ext_vector_type(8))) float v8f;

// Monotone float -> uint key so segment-max can use exact, order-invariant atomicMax.
__device__ __forceinline__ unsigned fkey(float f) {
  unsigned u = __float_as_uint(f);
  return (u & 0x80000000u) ? ~u : (u | 0x80000000u);
}
__device__ __forceinline__ float funkey(unsigned k) {
  return __uint_as_float((k & 0x80000000u) ? (k & 0x7fffffffu) : ~k);
}
__device__ __forceinline__ float lrelu(float x) { return x > 0.f ? x : NEG_SLOPE * x; }

__device__ __forceinline__ void edge_endpoints(const int* __restrict__ ei, int NE, int e,
                                               int& s, int& d) {
  if (e < NE) { s = ei[e]; d = ei[NE + e]; } else { s = d = e - NE; }  // self-loops appended
}

// ---------------- Layer-1 transform: h1[N,64] = x[N,2] @ W1[2,64] -----------------
// One 16-node tile per wave; K=2 padded to 4; 4 column tiles of 16.
// V_WMMA_F32_16X16X4_F32: A 16x4 f32 (lanes0-15 K=0,1; lanes16-31 K=2,3),
// B 4x16 f32 (N in lanes), C/D 16x16 f32 (8 VGPRs, M=r(+8), N=lane&15).
__global__ void k_l1_transform(const float* __restrict__ x, const float* __restrict__ W1,
                               float* __restrict__ h1, int nTiles, int N) {
  const int lane = threadIdx.x & 31;
  const int tile = blockIdx.x * (blockDim.x >> 5) + (threadIdx.x >> 5);
  if (tile >= nTiles) return;   // wave-uniform: EXEC stays all-1s for WMMA
  const int base = tile * 16;
  const int m  = lane & 15;     // A row / B,D column
  const int kk = lane >> 4;     // K sub-block: K = 2*kk + {0,1}
  const int row = base + m;
  const bool full = (base + 16 <= N);  // wave-uniform full-tile fast path
  v2f a; a[0] = 0.f; a[1] = 0.f;
  if (kk == 0 && row < N) { a[0] = x[row * 2 + 0]; a[1] = x[row * 2 + 1]; }  // K=2,3 pad = 0
#pragma unroll
  for (int jt = 0; jt < 4; ++jt) {
    const int j = jt * 16 + m;
    v2f b; b[0] = 0.f; b[1] = 0.f;
    if (kk == 0) { b[0] = W1[j]; b[1] = W1[64 + j]; }
    v8f c = {};
    c = __builtin_amdgcn_wmma_f32_16x16x4_f32(false, a, false, b, (short)0, c, false, false);
    float* hp = h1 + (size_t)(base + 8 * kk) * 64 + j;
    if (full) {
#pragma unroll
      for (int r = 0; r < 8; ++r) hp[(size_t)r * 64] = c[r];
    } else {
#pragma unroll
      for (int r = 0; r < 8; ++r)
        if (base + r + 8 * kk < N) hp[(size_t)r * 64] = c[r];
    }
  }
}

// ---------------- Layer-2 transform: h2t[N,2] = h1out[N,64] @ W2[64,2] -------------
// K=64 -> 16 chained WMMA f32 16x16x4 accumulations; only 2 of 16 output cols kept.
__global__ void k_l2_transform(const float* __restrict__ hin, const float* __restrict__ W2,
                               float* __restrict__ h2t, int nTiles, int N) {
  const int lane = threadIdx.x & 31;
  const int tile = blockIdx.x * (blockDim.x >> 5) + (threadIdx.x >> 5);
  if (tile >= nTiles) return;   // wave-uniform
  const int base = tile * 16;
  const int m  = lane & 15;
  const int kk = lane >> 4;
  const int row = base + m;
  const bool rok = row < N;
  const bool full = (base + 16 <= N);
  v8f c = {};
#pragma unroll
  for (int kc = 0; kc < 16; ++kc) {
    const int k0 = kc * 4 + 2 * kk;
    v2f a; a[0] = 0.f; a[1] = 0.f;
    if (rok) { a[0] = hin[(size_t)row * 64 + k0]; a[1] = hin[(size_t)row * 64 + k0 + 1]; }
    v2f b; b[0] = 0.f; b[1] = 0.f;
    if (m < 2) { b[0] = W2[k0 * 2 + m]; b[1] = W2[(k0 + 1) * 2 + m]; }
    c = __builtin_amdgcn_wmma_f32_16x16x4_f32(false, a, false, b, (short)0, c, false, false);
  }
  if (m < 2) {
    float* op = h2t + (size_t)(base + 8 * kk) * 2 + m;
    if (full) {
#pragma unroll
      for (int r = 0; r < 8; ++r) op[(size_t)r * 2] = c[r];
    } else {
#pragma unroll
      for (int r = 0; r < 8; ++r)
        if (base + r + 8 * kk < N) op[(size_t)r * 2] = c[r];
    }
  }
}

// ---------------- Per-(node,head) attention scalars, layer 1 ----------------------
__global__ void k_att1(const float* __restrict__ h1, const float* __restrict__ att_src,
                       const float* __restrict__ att_dst, float* __restrict__ asrc,
                       float* __restrict__ adst, int NH) {
  int t = blockIdx.x * blockDim.x + threadIdx.x;
  if (t >= NH) return;
  int n = t >> 3, h = t & 7;
  const float4* hp = reinterpret_cast<const float4*>(h1 + (size_t)n * 64 + h * 8);
  const float4* ap = reinterpret_cast<const float4*>(att_src + h * 8);
  const float4* dp = reinterpret_cast<const float4*>(att_dst + h * 8);
  float4 v0 = hp[0], v1 = hp[1], s0 = ap[0], s1 = ap[1], d0 = dp[0], d1 = dp[1];
  asrc[t] = v0.x*s0.x + v0.y*s0.y + v0.z*s0.z + v0.w*s0.w
          + v1.x*s1.x + v1.y*s1.y + v1.z*s1.z + v1.w*s1.w;
  adst[t] = v0.x*d0.x + v0.y*d0.y + v0.z*d0.z + v0.w*d0.w
          + v1.x*d1.x + v1.y*d1.y + v1.z*d1.z + v1.w*d1.w;
}

__global__ void k_att2(const float* __restrict__ h2t, const float* __restrict__ as2,
                       const float* __restrict__ ad2, float* __restrict__ asrc2,
                       float* __restrict__ adst2, int N) {
  int n = blockIdx.x * blockDim.x + threadIdx.x;
  if (n >= N) return;
  float v0 = h2t[n * 2 + 0], v1 = h2t[n * 2 + 1];
  asrc2[n] = v0 * as2[0] + v1 * as2[1];
  adst2[n] = v0 * ad2[0] + v1 * ad2[1];
}

// ======================= Layer-1 edge passes (H=8, vectorized) ====================
__global__ void k_edge_max8(const int* __restrict__ ei, int NE, int Etot,
                            const float* __restrict__ asrc, const float* __restrict__ adst,
                            unsigned* __restrict__ mkey) {
  int e = blockIdx.x * blockDim.x + threadIdx.x;
  if (e >= Etot) return;
  int s, d; edge_endpoints(ei, NE, e, s, d);
  const float4* sp = reinterpret_cast<const float4*>(asrc + (size_t)s * 8);
  const float4* dp = reinterpret_cast<const float4*>(adst + (size_t)d * 8);
  float4 s0 = sp[0], s1 = sp[1], d0 = dp[0], d1 = dp[1];
  float sc[8] = { s0.x + d0.x, s0.y + d0.y, s0.z + d0.z, s0.w + d0.w,
                  s1.x + d1.x, s1.y + d1.y, s1.z + d1.z, s1.w + d1.w };
#pragma unroll
  for (int h = 0; h < 8; ++h) atomicMax(&mkey[(size_t)d * 8 + h], fkey(lrelu(sc[h])));
}

__global__ void k_edge_exp8(const int* __restrict__ ei, int NE, int Etot,
                            const float* __restrict__ asrc, const float* __restrict__ adst,
                            const unsigned* __restrict__ mkey, float* __restrict__ denom,
                            float* __restrict__ alpha) {
  int e = blockIdx.x * blockDim.x + threadIdx.x;
  if (e >= Etot) return;
  int s, d; edge_endpoints(ei, NE, e, s, d);
  const float4* sp = reinterpret_cast<const float4*>(asrc + (size_t)s * 8);
  const float4* dp = reinterpret_cast<const float4*>(adst + (size_t)d * 8);
  const uint4*  mp = reinterpret_cast<const uint4*>(mkey + (size_t)d * 8);
  float4 s0 = sp[0], s1 = sp[1], d0 = dp[0], d1 = dp[1];
  uint4  m0 = mp[0], m1 = mp[1];
  float sc[8] = { s0.x + d0.x, s0.y + d0.y, s0.z + d0.z, s0.w + d0.w,
                  s1.x + d1.x, s1.y + d1.y, s1.z + d1.z, s1.w + d1.w };
  unsigned mk[8] = { m0.x, m0.y, m0.z, m0.w, m1.x, m1.y, m1.z, m1.w };
  float ex[8];
#pragma unroll
  for (int h = 0; h < 8; ++h) ex[h] = expf(lrelu(sc[h]) - funkey(mk[h]));
  float4* ap = reinterpret_cast<float4*>(alpha + (size_t)e * 8);
  ap[0] = make_float4(ex[0], ex[1], ex[2], ex[3]);   // unnormalized stash (b128 stores)
  ap[1] = make_float4(ex[4], ex[5], ex[6], ex[7]);
#pragma unroll
  for (int h = 0; h < 8; ++h) atomicAdd(&denom[(size_t)d * 8 + h], ex[h]);
}

__global__ void k_edge_agg8(const int* __restrict__ ei, int NE, int Etot,
                            const float* __restrict__ feat, const float* __restrict__ denom,
                            float* __restrict__ alpha, float* __restrict__ agg) {
  int e = blockIdx.x * blockDim.x + threadIdx.x;
  if (e >= Etot) return;
  int s, d; edge_endpoints(ei, NE, e, s, d);
  float4* ap = reinterpret_cast<float4*>(alpha + (size_t)e * 8);
  const float4* dn = reinterpret_cast<const float4*>(denom + (size_t)d * 8);
  float4 a0 = ap[0], a1 = ap[1], n0 = dn[0], n1 = dn[1];
  float al[8] = { a0.x / (n0.x + EPS_F), a0.y / (n0.y + EPS_F),
                  a0.z / (n0.z + EPS_F), a0.w / (n0.w + EPS_F),
                  a1.x / (n1.x + EPS_F), a1.y / (n1.y + EPS_F),
                  a1.z / (n1.z + EPS_F), a1.w / (n1.w + EPS_F) };
  ap[0] = make_float4(al[0], al[1], al[2], al[3]);   // final normalized alpha
  ap[1] = make_float4(al[4], al[5], al[6], al[7]);
#pragma unroll
  for (int h = 0; h < 8; ++h) {
    const float4* fp = reinterpret_cast<const float4*>(feat + (size_t)s * 64 + h * 8);
    float4 f0 = fp[0], f1 = fp[1];
    float* ag = agg + (size_t)d * 64 + h * 8;
    atomicAdd(ag + 0, f0.x * al[h]); atomicAdd(ag + 1, f0.y * al[h]);
    atomicAdd(ag + 2, f0.z * al[h]); atomicAdd(ag + 3, f0.w * al[h]);
    atomicAdd(ag + 4, f1.x * al[h]); atomicAdd(ag + 5, f1.y * al[h]);
    atomicAdd(ag + 6, f1.z * al[h]); atomicAdd(ag + 7, f1.w * al[h]);
  }
}

// ======================= Layer-2 edge passes (H=1, C=2, scalar) ===================
__global__ void k_edge_max1(const int* __restrict__ ei, int NE, int Etot,
                            const float* __restrict__ asrc, const float* __restrict__ adst,
                            unsigned* __restrict__ mkey) {
  int e = blockIdx.x * blockDim.x + threadIdx.x;
  if (e >= Etot) return;
  int s, d; edge_endpoints(ei, NE, e, s, d);
  atomicMax(&mkey[d], fkey(lrelu(asrc[s] + adst[d])));
}

__global__ void k_edge_exp1(const int* __restrict__ ei, int NE, int Etot,
                            const float* __restrict__ asrc, const float* __restrict__ adst,
                            const unsigned* __restrict__ mkey, float* __restrict__ denom,
                            float* __restrict__ alpha) {
  int e = blockIdx.x * blockDim.x + threadIdx.x;
  if (e >= Etot) return;
  int s, d; edge_endpoints(ei, NE, e, s, d);
  float ex = expf(lrelu(asrc[s] + adst[d]) - funkey(mkey[d]));
  atomicAdd(&denom[d], ex);
  alpha[e] = ex;
}

__global__ void k_edge_agg1(const int* __restrict__ ei, int NE, int Etot,
                            const float* __restrict__ feat, const float* __restrict__ denom,
                            float* __restrict__ alpha, float* __restrict__ agg) {
  int e = blockIdx.x * blockDim.x + threadIdx.x;
  if (e >= Etot) return;
  int s, d; edge_endpoints(ei, NE, e, s, d);
  float al = alpha[e] / (denom[d] + EPS_F);
  alpha[e] = al;
  atomicAdd(&agg[(size_t)d * 2 + 0], feat[(size_t)s * 2 + 0] * al);
  atomicAdd(&agg[(size_t)d * 2 + 1], feat[(size_t)s * 2 + 1] * al);
}

// ---------------- ELU + bias (in place), final bias add ---------------------------
__global__ void k_elu_bias(float* __restrict__ a, const float* __restrict__ bias, int total) {
  int t = blockIdx.x * blockDim.x + threadIdx.x;
  if (t >= total) return;
  float v = a[t] + bias[t & 63];
  a[t] = v > 0.f ? v : expm1f(v);
}

__global__ void k_out2(const float* __restrict__ agg2, const float* __restrict__ bias2,
                       float* __restrict__ out, int total) {
  int t = blockIdx.x * blockDim.x + threadIdx.x;
  if (t >= total) return;
  out[t] = agg2[t] + bias2[t & 1];
}

extern "C" void kernel_launch(void* const* d_in, const int* in_sizes, int n_in,
                              void* d_out, int out_size, void* d_ws, size_t ws_size,
                              hipStream_t stream) {
  const float* x   = (const float*)d_in[0];
  const int*   ei  = (const int*)d_in[1];
  const float* W1  = (const float*)d_in[2];
  const float* as1 = (const float*)d_in[3];
  const float* ad1 = (const float*)d_in[4];
  const float* b1  = (const float*)d_in[5];
  const float* W2  = (const float*)d_in[6];
  const float* as2 = (const float*)d_in[7];
  const float* ad2 = (const float*)d_in[8];
  const float* b2  = (const float*)d_in[9];
  float* out = (float*)d_out;

  const int N    = in_sizes[0] / 2;   // x is [N,2]
  const int NE   = in_sizes[1] / 2;   // edge_index is [2,NE]
  const int Etot = NE + N;            // + self-loops

  // Output layout: h2 [N,2] | alpha1 [Etot,8] | alpha2 [Etot,1]
  float* h2_out = out;
  float* alpha1 = out + (size_t)N * 2;
  float* alpha2 = alpha1 + (size_t)Etot * 8;

  // Workspace layout (floats): non-zeroed block, then one contiguous zeroed block.
  float* ws     = (float*)d_ws;
  float* h1     = ws;                        // N*64
  float* asrc1  = h1 + (size_t)N * 64;       // N*8
  float* adst1  = asrc1 + (size_t)N * 8;     // N*8
  float* h2t    = adst1 + (size_t)N * 8;     // N*2
  float* asrc2p = h2t + (size_t)N * 2;       // N
  float* adst2p = asrc2p + N;                // N
  float* z      = adst2p + N;                // ---- zeroed block (84*N floats) ----
  unsigned* mkey1 = (unsigned*)z;            // N*8
  float* denom1 = z + (size_t)N * 8;         // N*8
  float* agg1   = denom1 + (size_t)N * 8;    // N*64  (ELU'd in place -> layer-2 input)
  unsigned* mkey2 = (unsigned*)(agg1 + (size_t)N * 64);  // N
  float* denom2 = agg1 + (size_t)N * 64 + N; // N
  float* agg2   = denom2 + N;                // N*2

  hipMemsetAsync(z, 0, (size_t)N * 84 * sizeof(float), stream);

  const int B = 256;                          // 8 wave32s per block
  const int tiles = (N + 15) / 16;
  const int tgrid = (tiles + 7) / 8;
  auto cdiv = [](int a, int b) { return (a + b - 1) / b; };

  // Layer 1
  k_l1_transform<<<tgrid, B, 0, stream>>>(x, W1, h1, tiles, N);
  k_att1<<<cdiv(N * 8, B), B, 0, stream>>>(h1, as1, ad1, asrc1, adst1, N * 8);
  k_edge_max8<<<cdiv(Etot, B), B, 0, stream>>>(ei, NE, Etot, asrc1, adst1, mkey1);
  k_edge_exp8<<<cdiv(Etot, B), B, 0, stream>>>(ei, NE, Etot, asrc1, adst1, mkey1, denom1, alpha1);
  k_edge_agg8<<<cdiv(Etot, B), B, 0, stream>>>(ei, NE, Etot, h1, denom1, alpha1, agg1);
  k_elu_bias<<<cdiv(N * 64, B), B, 0, stream>>>(agg1, b1, N * 64);

  // Layer 2
  k_l2_transform<<<tgrid, B, 0, stream>>>(agg1, W2, h2t, tiles, N);
  k_att2<<<cdiv(N, B), B, 0, stream>>>(h2t, as2, ad2, asrc2p, adst2p, N);
  k_edge_max1<<<cdiv(Etot, B), B, 0, stream>>>(ei, NE, Etot, asrc2p, adst2p, mkey2);
  k_edge_exp1<<<cdiv(Etot, B), B, 0, stream>>>(ei, NE, Etot, asrc2p, adst2p, mkey2, denom2, alpha2);
  k_edge_agg1<<<cdiv(Etot, B), B, 0, stream>>>(ei, NE, Etot, h2t, denom2, alpha2, agg2);
  k_out2<<<cdiv(N * 2, B), B, 0, stream>>>(agg2, b2, h2_out, N * 2);
}